// IntraAgg_12567074308149
// MI455X (gfx1250) — compile-verified
//
#include <hip/hip_runtime.h>
#include <math.h>

typedef __attribute__((ext_vector_type(2))) float v2f;
typedef __attribute__((ext_vector_type(8))) float v8f;

__device__ __forceinline__ float wave_sum32(float v) {
#pragma unroll
  for (int m = 16; m >= 1; m >>= 1) v += __shfl_xor(v, m, 32);
  return v;
}

// ---------------------------------------------------------------------------
// Kernel 1: per-row neighbor gather + masked means + sigmoid scores.
// One wave32 per row b; lane == neighbor k (K == 32 exactly).
// ---------------------------------------------------------------------------
__global__ void __launch_bounds__(256)
agg_kernel(const int* __restrict__ nodes, const int* __restrict__ neigh,
           const float* __restrict__ valid, const float* __restrict__ samp,
           const float* __restrict__ choose,
           const float* __restrict__ feat, const float* __restrict__ pe,
           float* __restrict__ Xgg,   // [2B x 128] : rows b=[self|agg_all], B+b=[self|agg_all2]
           float* __restrict__ Xs,    // [B x 128]  : [self|agg_ms]
           float* __restrict__ Xd,    // [B x 64]   : self - agg_mn
           float* __restrict__ out_scores,  // [B x 32]
           float* __restrict__ out_raw,     // [B x 64]
           int B)
{
  __shared__ float cpe[8][64];
  const int wave = threadIdx.x >> 5;
  const int lane = threadIdx.x & 31;
  int b = blockIdx.x * 8 + wave;
  if (b >= B) b = B - 1;   // keep all waves alive for __syncthreads (grid is exact anyway)

  const int node  = nodes[b];
  const int myidx = neigh[b * 32 + lane];
  const float vm = valid[b * 32 + lane];
  const float sm = samp[b * 32 + lane];
  const float cm = choose[b * 32 + lane];

  // mask sums + fallback (jnp.where(sum>0, masked, valid))
  const float w2r = vm * sm;
  const float wsr = vm * cm;
  const float wnr = vm * (1.0f - cm);
  const float sv = wave_sum32(vm);
  const float s2 = wave_sum32(w2r);
  const float ss = wave_sum32(wsr);
  const float sn = wave_sum32(wnr);
  const float w2  = (s2 > 0.0f) ? w2r : vm;  const float d2 = (s2 > 0.0f) ? s2 : sv;
  const float wsc = (ss > 0.0f) ? wsr : vm;  const float dsv = (ss > 0.0f) ? ss : sv;
  const float wnc = (sn > 0.0f) ? wnr : vm;  const float dnv = (sn > 0.0f) ? sn : sv;
  const float invv = 1.0f / fmaxf(sv, 1.0f);
  const float inv2 = 1.0f / fmaxf(d2, 1.0f);
  const float invs = 1.0f / fmaxf(dsv, 1.0f);
  const float invn = 1.0f / fmaxf(dnv, 1.0f);

  // self feats + center pe (lane owns cols 2l, 2l+1)
  const float2 self2 = *(const float2*)(feat + (size_t)node * 64 + 2 * lane);
  const float2 c2    = *(const float2*)(pe   + (size_t)node * 64 + 2 * lane);
  cpe[wave][2 * lane]     = c2.x;
  cpe[wave][2 * lane + 1] = c2.y;
  __syncthreads();

  // sigmoid(center_pe . nb_pe[k]) * ms  -- lane-per-k, center broadcast from LDS
  float dot = 0.0f;
#pragma unroll
  for (int t = 0; t < 16; ++t) {
    const float4 p4 = *(const float4*)(pe + (size_t)myidx * 64 + 4 * t);
    const float4 c4 = *(const float4*)(&cpe[wave][4 * t]);
    dot = fmaf(p4.x, c4.x, dot);
    dot = fmaf(p4.y, c4.y, dot);
    dot = fmaf(p4.z, c4.z, dot);
    dot = fmaf(p4.w, c4.w, dot);
  }
  out_scores[(size_t)b * 32 + lane] = wsc / (1.0f + __expf(-dot));

  // masked means over neighbors -- loop k, lanes split the 64 columns (2 each)
  float2 av = {0.f, 0.f}, a2 = {0.f, 0.f}, as_ = {0.f, 0.f}, an = {0.f, 0.f};
  for (int k = 0; k < 32; ++k) {
    const int   ik  = __shfl(myidx, k, 32);
    const float wvk = __shfl(vm,  k, 32);
    const float w2k = __shfl(w2,  k, 32);
    const float wsk = __shfl(wsc, k, 32);
    const float wnk = __shfl(wnc, k, 32);
    const float2 f2 = *(const float2*)(feat + (size_t)ik * 64 + 2 * lane);
    av.x  = fmaf(wvk, f2.x, av.x);   av.y  = fmaf(wvk, f2.y, av.y);
    a2.x  = fmaf(w2k, f2.x, a2.x);   a2.y  = fmaf(w2k, f2.y, a2.y);
    as_.x = fmaf(wsk, f2.x, as_.x);  as_.y = fmaf(wsk, f2.y, as_.y);
    an.x  = fmaf(wnk, f2.x, an.x);   an.y  = fmaf(wnk, f2.y, an.y);
  }

  const size_t rb = (size_t)b;
  float2 t;
  // Xgg row b : [self | agg_all]
  *(float2*)(Xgg + rb * 128 + 2 * lane) = self2;
  t.x = av.x * invv; t.y = av.y * invv;
  *(float2*)(Xgg + rb * 128 + 64 + 2 * lane) = t;
  // Xgg row B+b : [self | agg_all2]
  *(float2*)(Xgg + ((size_t)B + rb) * 128 + 2 * lane) = self2;
  t.x = a2.x * inv2; t.y = a2.y * inv2;
  *(float2*)(Xgg + ((size_t)B + rb) * 128 + 64 + 2 * lane) = t;
  // Xs row b : [self | agg_ms]
  *(float2*)(Xs + rb * 128 + 2 * lane) = self2;
  t.x = as_.x * invs; t.y = as_.y * invs;
  *(float2*)(Xs + rb * 128 + 64 + 2 * lane) = t;
  // Xd row b : self - agg_mn
  t.x = self2.x - an.x * invn; t.y = self2.y - an.y * invn;
  *(float2*)(Xd + rb * 64 + 2 * lane) = t;
  // raw_feats
  *(float2*)(out_raw + rb * 64 + 2 * lane) = self2;
}

// ---------------------------------------------------------------------------
// Kernel 2: Y[rows x 128] = relu( [X0 | X1][rows x KD] @ W[KD x 128] )
// FP32 WMMA: V_WMMA_F32_16X16X4_F32, one 16x16 output tile per wave.
// Block: 256 threads = 8 waves -> 16 rows x 128 cols per block.
// ---------------------------------------------------------------------------
template <int KD>
__global__ void __launch_bounds__(256)
gemm_relu(const float* __restrict__ X0, int ld0,
          const float* __restrict__ X1, int ld1,
          const float* __restrict__ W,
          float* __restrict__ Y, int ldY, int rows)
{
  constexpr int LOG = (KD == 64) ? 6 : (KD == 128) ? 7 : 8;
  constexpr int LDSS = KD + 2;            // padded row stride (even -> b64 aligned)
  __shared__ float xs[16 * LDSS];

  const int tid = threadIdx.x;
  const int rowBase = blockIdx.x * 16;

  // cooperative X tile load (handles the column-concat via X1)
  for (int i = tid; i < 16 * KD; i += 256) {
    const int r = i >> LOG;
    const int c = i & (KD - 1);
    const int row = rowBase + r;
    float v = 0.0f;
    if (row < rows) {
      if (KD > 128 && c >= 128) v = X1[(size_t)row * ld1 + (c - 128)];
      else                      v = X0[(size_t)row * ld0 + c];
    }
    xs[r * LDSS + c] = v;
  }
  __syncthreads();

  const int wave = tid >> 5;
  const int lane = tid & 31;
  const int m    = lane & 15;
  const int hi   = lane >> 4;
  const int n    = (wave << 4) + m;       // output column 0..127

  v8f acc = {0.f, 0.f, 0.f, 0.f, 0.f, 0.f, 0.f, 0.f};
#pragma unroll
  for (int k0 = 0; k0 < KD; k0 += 4) {
    const int kk = k0 + 2 * hi;           // lanes 0-15: K={k0,k0+1}; 16-31: K={k0+2,k0+3}
    v2f a, bf;
    a.x  = xs[m * LDSS + kk];
    a.y  = xs[m * LDSS + kk + 1];
    bf.x = W[(size_t)kk * 128 + n];
    bf.y = W[(size_t)(kk + 1) * 128 + n];
    acc = __builtin_amdgcn_wmma_f32_16x16x4_f32(
        /*neg_a=*/false, a, /*neg_b=*/false, bf,
        /*c_mod=*/(short)0, acc, /*reuse_a=*/false, /*reuse_b=*/false);
  }

#pragma unroll
  for (int v = 0; v < 8; ++v) {           // C/D layout: VGPR v -> rows v (lo) / v+8 (hi)
    const int row = rowBase + v + 8 * hi;
    if (row < rows) {
      const float x = acc[v];
      Y[(size_t)row * ldY + n] = x > 0.0f ? x : 0.0f;
    }
  }
}

// ---------------------------------------------------------------------------
extern "C" void kernel_launch(void* const* d_in, const int* in_sizes, int n_in,
                              void* d_out, int out_size, void* d_ws, size_t ws_size,
                              hipStream_t stream) {
  const int*   nodes  = (const int*)d_in[0];
  const int*   neigh  = (const int*)d_in[1];
  const float* valid  = (const float*)d_in[2];
  const float* samp   = (const float*)d_in[3];
  const float* choose = (const float*)d_in[4];
  const float* feat   = (const float*)d_in[5];
  const float* pe     = (const float*)d_in[6];
  const float* W      = (const float*)d_in[7];   // [128x128]
  const float* W1     = (const float*)d_in[8];   // [256x128]
  const float* W2     = (const float*)d_in[9];   // [64x128]
  const float* W5     = (const float*)d_in[10];  // [256x128]
  const float* Wg     = (const float*)d_in[11];  // [128x128]
  const int B = in_sizes[0];

  // d_out regions in reference return order
  float* out      = (float*)d_out;
  float* to_feats = out;                             // B*128
  float* scores   = to_feats + (size_t)B * 128;      // B*32
  float* gen      = scores   + (size_t)B * 32;       // B*128 (gen) + B*128 (gen2), contiguous
  float* gen2     = gen      + (size_t)B * 128;
  float* raw      = gen2     + (size_t)B * 128;      // B*64
  float* htop     = raw      + (size_t)B * 64;       // B*128
  (void)gen2; (void)out_size; (void)n_in; (void)ws_size;

  // workspace: Xgg [2B x 128] (reused later as LH [B x 256]), Xs [B x 128], Xd [B x 64]
  float* ws  = (float*)d_ws;
  float* Xgg = ws;
  float* Xs  = ws + (size_t)B * 256;
  float* Xd  = Xs + (size_t)B * 128;
  float* LH  = Xgg;   // alias: Xgg fully consumed before LH is written (stream order)

  agg_kernel<<<(B + 7) / 8, 256, 0, stream>>>(
      nodes, neigh, valid, samp, choose, feat, pe,
      Xgg, Xs, Xd, scores, raw, B);

  // gen_feats + gen2_feats in one GEMM (rows 0..B-1 and B..2B-1 share Wg)
  gemm_relu<128><<<(2 * B + 15) / 16, 256, 0, stream>>>(
      Xgg, 128, nullptr, 0, Wg, gen, 128, 2 * B);
  // l_f -> LH cols 0..127
  gemm_relu<128><<<(B + 15) / 16, 256, 0, stream>>>(
      Xs, 128, nullptr, 0, W, LH, 256, B);
  // h_f -> LH cols 128..255
  gemm_relu<64><<<(B + 15) / 16, 256, 0, stream>>>(
      Xd, 64, nullptr, 0, W2, LH + 128, 256, B);
  // h_top = relu([l_f|h_f] @ W5)
  gemm_relu<256><<<(B + 15) / 16, 256, 0, stream>>>(
      LH, 256, LH + 128, 256, W5, htop, 128, B);
  // to_feats = relu([h_top|gen] @ W1)
  gemm_relu<256><<<(B + 15) / 16, 256, 0, stream>>>(
      htop, 128, gen, 128, W1, to_feats, 128, B);
}